// TemporalGCN_31636729102359
// MI455X (gfx1250) — compile-verified
//
#include <hip/hip_runtime.h>

// Problem constants (match reference)
#define TT 8
#define NN 50000
#define EE 800000
#define FF 128
#define HH 128

typedef float v2f __attribute__((ext_vector_type(2)));
typedef float v8f __attribute__((ext_vector_type(8)));

// ---------------------------------------------------------------------------
// WMMA fp32 GEMM: C[N,128] = A[N,128] @ W[128,128] (+ bias)
// One wave -> one 16x16 C tile. 8 waves/block cover 16 rows x 128 cols.
// K loop: 32 x v_wmma_f32_16x16x4_f32 (fully unrolled).
// A frag (16x4 f32): lane<16: M=lane, K={k,k+1}; lane>=16: M=lane-16, K={k+2,k+3}
// B frag (4x16 f32): v0 = row k+2*hi, v1 = row k+2*hi+1, col = lane%16
// D (16x16 f32):     acc[i] -> row (i + 8*hi), col lane%16
// ---------------------------------------------------------------------------
__global__ __launch_bounds__(256) void wmma_gemm128_kernel(
    const float* __restrict__ A, const float* __restrict__ W,
    const float* __restrict__ bias, float* __restrict__ C, int N) {
  const int wave = threadIdx.x >> 5;
  const int lane = threadIdx.x & 31;
  const int lr   = lane & 15;   // row (A) / col (B,D) within tile
  const int hi   = lane >> 4;   // lane half select
  const int row0 = blockIdx.x * 16;
  const int col0 = wave * 16;

  const float* arow = A + (size_t)(row0 + lr) * 128;
  v8f acc = {};
#pragma unroll
  for (int k = 0; k < 128; k += 4) {
    v2f a = *(const v2f*)(arow + k + 2 * hi);            // contiguous K pair
    v2f b;
    b.x = W[(size_t)(k + 2 * hi) * 128 + col0 + lr];
    b.y = W[(size_t)(k + 2 * hi + 1) * 128 + col0 + lr];
    acc = __builtin_amdgcn_wmma_f32_16x16x4_f32(
        /*neg_a=*/false, a, /*neg_b=*/false, b,
        /*c_mod=*/(short)0, acc, /*reuse_a=*/false, /*reuse_b=*/false);
  }
  const float bv = bias ? bias[col0 + lr] : 0.0f;
#pragma unroll
  for (int i = 0; i < 8; ++i) {
    const int r = row0 + i + 8 * hi;
    C[(size_t)r * 128 + col0 + lr] = acc[i] + bv;
  }
}

// ---------------------------------------------------------------------------
// Degree / normalization helpers
// ---------------------------------------------------------------------------
__global__ void fill_kernel(float* __restrict__ p, float v, int n) {
  int i = blockIdx.x * blockDim.x + threadIdx.x;
  if (i < n) p[i] = v;
}

__global__ void deg_accum_kernel(const int* __restrict__ dst,
                                 float* __restrict__ deg, int E) {
  int e = blockIdx.x * blockDim.x + threadIdx.x;
  if (e < E) atomicAdd(&deg[dst[e]], 1.0f);
}

__global__ void rsqrt_kernel(float* __restrict__ p, int n) {
  int i = blockIdx.x * blockDim.x + threadIdx.x;
  if (i < n) p[i] = rsqrtf(p[i]);
}

// agg[n][:] = hw[n][:] * dis[n]^2 + conv_b   (self-loop term; doubles as init)
__global__ __launch_bounds__(256) void init_agg_kernel(
    const float* __restrict__ hw, const float* __restrict__ dis,
    const float* __restrict__ cb, float* __restrict__ agg, int N) {
  int idx = blockIdx.x * blockDim.x + threadIdx.x;   // N*32 threads, float4 each
  if (idx >= N * 32) return;
  int n = idx >> 5, q = idx & 31;
  float sw = dis[n];
  sw *= sw;
  float4 v = ((const float4*)(hw + (size_t)n * 128))[q];
  float4 b = ((const float4*)cb)[q];
  float4 r = {v.x * sw + b.x, v.y * sw + b.y, v.z * sw + b.z, v.w * sw + b.w};
  ((float4*)(agg + (size_t)n * 128))[q] = r;
}

// One wave per edge: coalesced float4 gather of hw[src], 4 atomic f32 adds/lane
__global__ __launch_bounds__(256) void scatter_kernel(
    const float* __restrict__ hw, const int* __restrict__ src,
    const int* __restrict__ dst, const float* __restrict__ dis,
    float* __restrict__ agg, int E) {
  int e = blockIdx.x * 8 + (threadIdx.x >> 5);
  if (e >= E) return;
  int lane = threadIdx.x & 31;
  int s = src[e], d = dst[e];
  float w = dis[s] * dis[d];
  float4 v = ((const float4*)(hw + (size_t)s * 128))[lane];
  float* ad = agg + (size_t)d * 128 + lane * 4;
  atomicAdd(ad + 0, v.x * w);
  atomicAdd(ad + 1, v.y * w);
  atomicAdd(ad + 2, v.z * w);
  atomicAdd(ad + 3, v.w * w);
}

// LayerNorm + ReLU (+ residual): one wave32 per node, shfl_xor reductions
__global__ __launch_bounds__(256) void ln_relu_kernel(
    const float* __restrict__ agg, const float* __restrict__ g,
    const float* __restrict__ b, const float* __restrict__ h_prev,
    float* __restrict__ h_out, int N, int use_res) {
  int node = blockIdx.x * 8 + (threadIdx.x >> 5);
  if (node >= N) return;
  int lane = threadIdx.x & 31;
  const float* row = agg + (size_t)node * 128;
  float vals[4];
  float s = 0.0f;
#pragma unroll
  for (int i = 0; i < 4; ++i) {
    vals[i] = row[lane + 32 * i];
    s += vals[i];
  }
#pragma unroll
  for (int off = 16; off > 0; off >>= 1) s += __shfl_xor(s, off, 32);
  float mean = s * (1.0f / 128.0f);
  float var = 0.0f;
#pragma unroll
  for (int i = 0; i < 4; ++i) {
    float d = vals[i] - mean;
    var += d * d;
  }
#pragma unroll
  for (int off = 16; off > 0; off >>= 1) var += __shfl_xor(var, off, 32);
  float inv = rsqrtf(var * (1.0f / 128.0f) + 1e-5f);
#pragma unroll
  for (int i = 0; i < 4; ++i) {
    int c = lane + 32 * i;
    float y = (vals[i] - mean) * inv * g[c] + b[c];
    y = fmaxf(y, 0.0f);
    if (use_res) y += h_prev[(size_t)node * 128 + c];
    h_out[(size_t)node * 128 + c] = y;
  }
}

// Global mean pool: 128 threads/block = columns; blocks stride node range
__global__ __launch_bounds__(128) void pool_kernel(const float* __restrict__ h,
                                                   float* __restrict__ emb,
                                                   int N, float scale) {
  int c = threadIdx.x;
  float s = 0.0f;
  for (int n = blockIdx.x; n < N; n += gridDim.x) s += h[(size_t)n * 128 + c];
  atomicAdd(&emb[c], s * scale);
}

// ---------------------------------------------------------------------------
// 2-layer LSTM over T=8 + head MLP, single block (tiny workload)
// ---------------------------------------------------------------------------
__global__ __launch_bounds__(512) void lstm_head_kernel(
    const float* __restrict__ embs, const float* __restrict__ Wih,
    const float* __restrict__ Whh, const float* __restrict__ bih,
    const float* __restrict__ bhh, const float* __restrict__ W1,
    const float* __restrict__ b1, const float* __restrict__ W2,
    const float* __restrict__ b2, float* __restrict__ out) {
  __shared__ float xbuf[TT][HH];
  __shared__ float ybuf[TT][HH];
  __shared__ float hst[HH], cst[HH];
  __shared__ float gates[4 * HH];
  __shared__ float hid[HH / 2];
  int tid = threadIdx.x;  // 512
  for (int i = tid; i < TT * HH; i += 512) ((float*)xbuf)[i] = embs[i];
  __syncthreads();
  for (int l = 0; l < 2; ++l) {
    const float* wih = Wih + (size_t)l * 4 * HH * HH;
    const float* whh = Whh + (size_t)l * 4 * HH * HH;
    const float* bi = bih + l * 4 * HH;
    const float* bh = bhh + l * 4 * HH;
    if (tid < HH) { hst[tid] = 0.0f; cst[tid] = 0.0f; }
    __syncthreads();
    for (int t = 0; t < TT; ++t) {
      float acc = bi[tid] + bh[tid];
      const float* wr = wih + (size_t)tid * HH;
      const float* ur = whh + (size_t)tid * HH;
#pragma unroll 4
      for (int k = 0; k < HH; ++k) acc += wr[k] * xbuf[t][k] + ur[k] * hst[k];
      gates[tid] = acc;
      __syncthreads();
      if (tid < HH) {
        float ig = 1.0f / (1.0f + expf(-gates[tid]));
        float fg = 1.0f / (1.0f + expf(-gates[HH + tid]));
        float gg = tanhf(gates[2 * HH + tid]);
        float og = 1.0f / (1.0f + expf(-gates[3 * HH + tid]));
        float c = fg * cst[tid] + ig * gg;
        float hh = og * tanhf(c);
        cst[tid] = c;
        hst[tid] = hh;
        ybuf[t][tid] = hh;
      }
      __syncthreads();
    }
    for (int i = tid; i < TT * HH; i += 512) ((float*)xbuf)[i] = ((float*)ybuf)[i];
    __syncthreads();
  }
  // head: relu(final @ W1 + b1) @ W2 + b2 ; final = xbuf[T-1]
  if (tid < HH / 2) {
    float a = b1[tid];
    for (int k = 0; k < HH; ++k) a += xbuf[TT - 1][k] * W1[k * (HH / 2) + tid];
    hid[tid] = fmaxf(a, 0.0f);
  }
  __syncthreads();
  if (tid == 0) {
    float p = b2[0];
    for (int j = 0; j < HH / 2; ++j) p += hid[j] * W2[j];
    out[0] = p;
  }
  if (tid < HH) out[1 + tid] = xbuf[TT - 1][tid];
}

// ---------------------------------------------------------------------------
extern "C" void kernel_launch(void* const* d_in, const int* in_sizes, int n_in,
                              void* d_out, int out_size, void* d_ws,
                              size_t ws_size, hipStream_t stream) {
  const float* x_seq   = (const float*)d_in[0];   // [T,N,F]
  const int*   eidx    = (const int*)d_in[1];     // [T,2,E]
  const float* W_in    = (const float*)d_in[2];   // [F,H]
  const float* b_in    = (const float*)d_in[3];   // [H]
  const float* conv_W  = (const float*)d_in[4];   // [3,H,H]
  const float* conv_b  = (const float*)d_in[5];   // [3,H]
  const float* ln_g    = (const float*)d_in[6];   // [3,H]
  const float* ln_b    = (const float*)d_in[7];   // [3,H]
  const float* Wih     = (const float*)d_in[8];   // [2,4H,H]
  const float* Whh     = (const float*)d_in[9];   // [2,4H,H]
  const float* bih     = (const float*)d_in[10];  // [2,4H]
  const float* bhh     = (const float*)d_in[11];  // [2,4H]
  const float* W1      = (const float*)d_in[12];  // [H,H/2]
  const float* b1      = (const float*)d_in[13];
  const float* W2      = (const float*)d_in[14];  // [H/2,1]
  const float* b2      = (const float*)d_in[15];
  float* out = (float*)d_out;                     // [1 + H]

  // Workspace layout (floats)
  float* ws   = (float*)d_ws;
  float* dis  = ws;                                // N
  float* hA   = dis + NN;                          // N*H
  float* hB   = hA + (size_t)NN * HH;              // N*H
  float* hw   = hB + (size_t)NN * HH;              // N*H
  float* agg  = hw + (size_t)NN * HH;              // N*H
  float* embs = agg + (size_t)NN * HH;             // T*H

  fill_kernel<<<(TT * HH + 255) / 256, 256, 0, stream>>>(embs, 0.0f, TT * HH);

  for (int t = 0; t < TT; ++t) {
    const float* x_t = x_seq + (size_t)t * NN * FF;
    const int* src = eidx + (size_t)t * 2 * EE;
    const int* dst = src + EE;

    // D^{-1/2} with self loops
    fill_kernel<<<(NN + 255) / 256, 256, 0, stream>>>(dis, 1.0f, NN);
    deg_accum_kernel<<<(EE + 255) / 256, 256, 0, stream>>>(dst, dis, EE);
    rsqrt_kernel<<<(NN + 255) / 256, 256, 0, stream>>>(dis, NN);

    // input projection (WMMA)
    wmma_gemm128_kernel<<<NN / 16, 256, 0, stream>>>(x_t, W_in, b_in, hA, NN);

    float* hcur = hA;
    float* hnext = hB;
    for (int l = 0; l < 3; ++l) {
      wmma_gemm128_kernel<<<NN / 16, 256, 0, stream>>>(
          hcur, conv_W + (size_t)l * HH * HH, nullptr, hw, NN);
      init_agg_kernel<<<(NN * 32 + 255) / 256, 256, 0, stream>>>(
          hw, dis, conv_b + l * HH, agg, NN);
      scatter_kernel<<<(EE + 7) / 8, 256, 0, stream>>>(hw, src, dst, dis, agg, EE);
      ln_relu_kernel<<<(NN + 7) / 8, 256, 0, stream>>>(
          agg, ln_g + l * HH, ln_b + l * HH, hcur, hnext, NN, l > 0 ? 1 : 0);
      float* tmp = hcur; hcur = hnext; hnext = tmp;
    }
    pool_kernel<<<256, 128, 0, stream>>>(hcur, embs + t * HH, NN, 1.0f / NN);
  }

  lstm_head_kernel<<<1, 512, 0, stream>>>(embs, Wih, Whh, bih, bhh, W1, b1, W2,
                                          b2, out);
}